// DA_84825604096359
// MI455X (gfx1250) — compile-verified
//
#include <hip/hip_runtime.h>
#include <math.h>

// MI455X / gfx1250, wave32. Memory-bound strip-pooling attention:
//   pass1: per-(b,c) row/col/global pooled stats (reads x once, 128 MiB)
//   pass2: tiny strip convs + BN + sigmoid + gate means on (B,C,128)
//   pass3: 2x2 softmax mix per (b,c); channel-attention MLP via
//          V_WMMA_F32_16X16X4_F32 (fp32 matrix core path)
//   pass4: fused elementwise out = x*(1 + a*lam_h*w_h + a*lam_w*w_w + b*ca_w)
// HBM traffic ~256 MiB (x stays in 192MB L2 between passes) -> ~11us roofline.

#define BATCH 8
#define CH 256
#define HH 128
#define WWD 128
#define HWC (HH*WWD)
#define NBC (BATCH*CH)
#define EPSV 1e-5f

typedef float v2f __attribute__((ext_vector_type(2)));
typedef float v8f __attribute__((ext_vector_type(8)));

__device__ __forceinline__ float wave_sum(float v) {
#pragma unroll
  for (int o = 16; o > 0; o >>= 1) v += __shfl_xor(v, o, 32);
  return v;
}
__device__ __forceinline__ float wave_max(float v) {
#pragma unroll
  for (int o = 16; o > 0; o >>= 1) v = fmaxf(v, __shfl_xor(v, o, 32));
  return v;
}

// ---------------- Pass 1: pooled statistics -------------------------------
// One block per (b,c) 128x128 tile. float4 loads, flat index f=(i*256+tid)*4:
//   h = 8*i + (tid>>5)  (whole wave shares a row -> shfl row reduce, no atomics)
//   w = 4*(tid&31)..+3  (fixed per lane -> column partials in registers)
__global__ __launch_bounds__(256) void pool_kernel(
    const float* __restrict__ x,
    const float* __restrict__ hw_sq_w, const float* __restrict__ hw_sq_b,
    const float* __restrict__ ww_sq_w, const float* __restrict__ ww_sq_b,
    float* __restrict__ s_h, float* __restrict__ s_w,
    float* __restrict__ mean_bc, float* __restrict__ max_bc)
{
  __shared__ float rowsum[HH], rowmax[HH];
  __shared__ float colps[8][WWD], colpm[8][WWD];
  __shared__ float gs[8], gm[8];
  const int bc = blockIdx.x;
  const int tid = threadIdx.x;
  const int lane = tid & 31, wave = tid >> 5;
  const float4* xt = (const float4*)(x + (size_t)bc * HWC);

  float cs0 = 0.f, cs1 = 0.f, cs2 = 0.f, cs3 = 0.f;
  float cm0 = -INFINITY, cm1 = -INFINITY, cm2 = -INFINITY, cm3 = -INFINITY;
  float gsum = 0.f, gmx = -INFINITY;
#pragma unroll 4
  for (int i = 0; i < 16; ++i) {
    float4 v = xt[i * 256 + tid];
    float rs = v.x + v.y + v.z + v.w;
    float rm = fmaxf(fmaxf(v.x, v.y), fmaxf(v.z, v.w));
    gsum += rs;
    gmx = fmaxf(gmx, rm);
    float wrs = wave_sum(rs);
    float wrm = wave_max(rm);
    if (lane == 0) { rowsum[i * 8 + wave] = wrs; rowmax[i * 8 + wave] = wrm; }
    cs0 += v.x; cs1 += v.y; cs2 += v.z; cs3 += v.w;
    cm0 = fmaxf(cm0, v.x); cm1 = fmaxf(cm1, v.y);
    cm2 = fmaxf(cm2, v.z); cm3 = fmaxf(cm3, v.w);
  }
  const int w0 = lane * 4;
  colps[wave][w0 + 0] = cs0; colps[wave][w0 + 1] = cs1;
  colps[wave][w0 + 2] = cs2; colps[wave][w0 + 3] = cs3;
  colpm[wave][w0 + 0] = cm0; colpm[wave][w0 + 1] = cm1;
  colpm[wave][w0 + 2] = cm2; colpm[wave][w0 + 3] = cm3;
  gsum = wave_sum(gsum);
  gmx = wave_max(gmx);
  if (lane == 0) { gs[wave] = gsum; gm[wave] = gmx; }
  __syncthreads();

  if (tid < HH) {
    // height-direction strip input: pool over W (row stats)
    float ravg = rowsum[tid] * (1.f / WWD);
    s_h[bc * HH + tid] = hw_sq_w[0] * rowmax[tid] + hw_sq_w[1] * ravg + hw_sq_b[0];
    // width-direction strip input: pool over H (column stats)
    float csum = 0.f, cmax = -INFINITY;
#pragma unroll
    for (int j = 0; j < 8; ++j) { csum += colps[j][tid]; cmax = fmaxf(cmax, colpm[j][tid]); }
    float cavg = csum * (1.f / HH);
    s_w[bc * WWD + tid] = ww_sq_w[0] * cmax + ww_sq_w[1] * cavg + ww_sq_b[0];
  }
  if (tid == 0) {
    float ts = 0.f, tm = -INFINITY;
#pragma unroll
    for (int j = 0; j < 8; ++j) { ts += gs[j]; tm = fmaxf(tm, gm[j]); }
    mean_bc[bc] = ts * (1.f / HWC);
    max_bc[bc] = tm;
  }
}

// ---------------- Pass 2: dilated strip convs + BN + sigmoid + gate -------
// One block per (b,c); threads 0..127 do the H strip, 128..255 the W strip.
__global__ __launch_bounds__(256) void strip_kernel(
    const float* __restrict__ s_h, const float* __restrict__ s_w,
    const float* __restrict__ hw_conv, const float* __restrict__ ww_conv,
    const float* __restrict__ hw_g, const float* __restrict__ hw_b,
    const float* __restrict__ hw_m, const float* __restrict__ hw_v,
    const float* __restrict__ ww_g, const float* __restrict__ ww_b,
    const float* __restrict__ ww_m, const float* __restrict__ ww_v,
    const float* __restrict__ gate_w,
    const float* __restrict__ g_g, const float* __restrict__ g_b,
    const float* __restrict__ g_m, const float* __restrict__ g_v,
    float* __restrict__ w_h, float* __restrict__ w_w,
    float* __restrict__ gh, float* __restrict__ gw)
{
  __shared__ float sbuf[2][128];
  __shared__ float gred[8];
  const int bc = blockIdx.x, c = bc & (CH - 1);
  const int tid = threadIdx.x;
  const int dir = tid >> 7;      // 0: height strip, 1: width strip
  const int l = tid & 127;
  sbuf[dir][l] = dir ? s_w[bc * 128 + l] : s_h[bc * 128 + l];
  __syncthreads();

  const float* s = sbuf[dir];
  const float* cw = (dir ? ww_conv : hw_conv) + c * 7;  // (3,C,1,7) layout
  float y = 0.f;
#pragma unroll
  for (int d = 0; d < 3; ++d) {
    const int dil = d + 1;
#pragma unroll
    for (int k = 0; k < 7; ++k) {
      const int idx = l + (k - 3) * dil;
      const float sv = (idx >= 0 && idx < 128) ? s[idx] : 0.f;
      y += cw[d * (CH * 7) + k] * sv;
    }
  }
  const float bg = dir ? ww_g[c] : hw_g[c];
  const float bb = dir ? ww_b[c] : hw_b[c];
  const float bm = dir ? ww_m[c] : hw_m[c];
  const float bv = dir ? ww_v[c] : hw_v[c];
  y = (y - bm) * (bg * rsqrtf(bv + EPSV)) + bb;
  const float wv = 1.f / (1.f + expf(-y));
  (dir ? w_w : w_h)[bc * 128 + l] = wv;

  // gate(seq) = mean_l relu(BN(seq * gate_w))
  float t = (wv * gate_w[c] - g_m[c]) * (g_g[c] * rsqrtf(g_v[c] + EPSV)) + g_b[c];
  t = fmaxf(t, 0.f);
  const float wsum = wave_sum(t);
  const int lane = tid & 31, wave = tid >> 5;
  if (lane == 0) gred[wave] = wsum;
  __syncthreads();
  if (tid == 0)   gh[bc] = (gred[0] + gred[1] + gred[2] + gred[3]) * (1.f / 128.f);
  if (tid == 128) gw[bc] = (gred[4] + gred[5] + gred[6] + gred[7]) * (1.f / 128.f);
}

// ---------------- Pass 3a: orientation softmax mix ------------------------
__global__ __launch_bounds__(256) void lam_kernel(
    const float* __restrict__ gh, const float* __restrict__ gw,
    const float* __restrict__ mixW, const float* __restrict__ mixb,
    const float* __restrict__ alpha_p,
    float* __restrict__ a_lam_h, float* __restrict__ a_lam_w)
{
  const int i = blockIdx.x * blockDim.x + threadIdx.x;
  if (i >= NBC) return;
  const float g0 = gh[i], g1 = gw[i];
  const float z0 = mixW[0] * g0 + mixW[1] * g1 + mixb[0];
  const float z1 = mixW[2] * g0 + mixW[3] * g1 + mixb[1];
  const float m = fmaxf(z0, z1);
  const float e0 = expf(z0 - m), e1 = expf(z1 - m);
  const float inv = alpha_p[0] / (e0 + e1);
  a_lam_h[i] = e0 * inv;
  a_lam_w[i] = e1 * inv;
}

// ---------------- Pass 3b: channel-attention MLP via fp32 WMMA ------------
// Single wave32, EXEC all-ones around the WMMAs.
// A(16x256): rows 0..7 = mean(b), rows 8..15 = max(b).
// Hd(16x16) = A @ fc1^T  (64 x V_WMMA_F32_16X16X4_F32)
// fc is linear after relu -> hsum[b][j] = relu(Hd[b]) + relu(Hd[b+8])
// out(8x256) = hsum @ fc2^T (16 N-tiles x 4 k-steps)
// A frag (16x4 f32, ISA 7.12.2): lanes0-15 K={0,1}, lanes16-31 K={2,3}.
// D (16x16 f32): vgpr g holds rows g (lanes0-15) and g+8 (lanes16-31).
__global__ __launch_bounds__(32) void ca_kernel(
    const float* __restrict__ mean_bc, const float* __restrict__ max_bc,
    const float* __restrict__ fc1, const float* __restrict__ fc2,
    const float* __restrict__ beta_p, float* __restrict__ comb)
{
  __shared__ float hsum[8 * 16];
  const int lane = threadIdx.x;
  const int r = lane & 15;
  const int koff = (lane >> 4) * 2;
  const float* arow = (r < 8) ? (mean_bc + r * CH) : (max_bc + (r - 8) * CH);

  v8f acc = {0.f, 0.f, 0.f, 0.f, 0.f, 0.f, 0.f, 0.f};
  for (int k0 = 0; k0 < CH; k0 += 4) {
    const int ka = k0 + koff;
    v2f a; a.x = arow[ka]; a.y = arow[ka + 1];
    v2f b; b.x = fc1[r * CH + ka]; b.y = fc1[r * CH + ka + 1];  // B[k][n]=fc1[n][k]
    acc = __builtin_amdgcn_wmma_f32_16x16x4_f32(false, a, false, b,
                                                (short)0, acc, false, false);
  }
#pragma unroll
  for (int m = 0; m < 8; ++m) {
    float v = fmaxf(acc[m], 0.f);
    v += __shfl_xor(v, 16, 32);         // fold relu(mean half)+relu(max half)
    if (lane < 16) hsum[m * 16 + lane] = v;
  }
  __syncthreads();
  const float beta = beta_p[0];
  for (int t = 0; t < 16; ++t) {
    v8f cacc = {0.f, 0.f, 0.f, 0.f, 0.f, 0.f, 0.f, 0.f};
#pragma unroll
    for (int kk = 0; kk < 16; kk += 4) {
      const int ka = kk + koff;
      v2f a;
      a.x = (r < 8) ? hsum[r * 16 + ka] : 0.f;       // pad rows 8..15 with 0
      a.y = (r < 8) ? hsum[r * 16 + ka + 1] : 0.f;
      const int n = t * 16 + r;
      v2f b; b.x = fc2[n * 16 + ka]; b.y = fc2[n * 16 + ka + 1];  // B[k][n]=fc2[n][k]
      cacc = __builtin_amdgcn_wmma_f32_16x16x4_f32(false, a, false, b,
                                                   (short)0, cacc, false, false);
    }
    if (lane < 16) {
#pragma unroll
      for (int m = 0; m < 8; ++m) {
        const float caw = 1.f / (1.f + expf(-cacc[m]));
        comb[m * CH + t * 16 + lane] = 1.f + beta * caw;  // 1 + beta*ca_w
      }
    }
  }
}

// ---------------- Pass 4: fused elementwise apply -------------------------
// out = x * (comb[bc] + alpha*lam_h*w_h[h] + alpha*lam_w*w_w[w])
__global__ __launch_bounds__(256) void apply_kernel(
    const float* __restrict__ x,
    const float* __restrict__ w_h, const float* __restrict__ w_w,
    const float* __restrict__ a_lam_h, const float* __restrict__ a_lam_w,
    const float* __restrict__ comb, float* __restrict__ out)
{
  __shared__ float hterm[HH];
  const int bc = blockIdx.x, tid = threadIdx.x;
  const float alh = a_lam_h[bc], alw = a_lam_w[bc], cmb = comb[bc];
  if (tid < HH) hterm[tid] = cmb + alh * w_h[bc * HH + tid];
  const int lane = tid & 31;
  const float4* wwr = (const float4*)(w_w + bc * WWD);
  float4 w4 = wwr[lane];               // lane's fixed columns 4*lane..+3
  w4.x *= alw; w4.y *= alw; w4.z *= alw; w4.w *= alw;
  __syncthreads();
  const float4* xt = (const float4*)(x + (size_t)bc * HWC);
  float4* ot = (float4*)(out + (size_t)bc * HWC);
#pragma unroll 4
  for (int i = 0; i < 16; ++i) {
    float4 v = xt[i * 256 + tid];
    const float b = hterm[i * 8 + (tid >> 5)];
    float4 o;
    o.x = v.x * (b + w4.x);
    o.y = v.y * (b + w4.y);
    o.z = v.z * (b + w4.z);
    o.w = v.w * (b + w4.w);
    ot[i * 256 + tid] = o;
  }
}

extern "C" void kernel_launch(void* const* d_in, const int* in_sizes, int n_in,
                              void* d_out, int out_size, void* d_ws, size_t ws_size,
                              hipStream_t stream)
{
  const float* x       = (const float*)d_in[0];
  const float* hw_sq_w = (const float*)d_in[1];
  const float* hw_sq_b = (const float*)d_in[2];
  const float* hw_conv = (const float*)d_in[3];
  const float* hw_g    = (const float*)d_in[4];
  const float* hw_b    = (const float*)d_in[5];
  const float* hw_m    = (const float*)d_in[6];
  const float* hw_v    = (const float*)d_in[7];
  const float* ww_sq_w = (const float*)d_in[8];
  const float* ww_sq_b = (const float*)d_in[9];
  const float* ww_conv = (const float*)d_in[10];
  const float* ww_g    = (const float*)d_in[11];
  const float* ww_b    = (const float*)d_in[12];
  const float* ww_m    = (const float*)d_in[13];
  const float* ww_v    = (const float*)d_in[14];
  const float* gate_w  = (const float*)d_in[15];
  const float* g_g     = (const float*)d_in[16];
  const float* g_b     = (const float*)d_in[17];
  const float* g_m     = (const float*)d_in[18];
  const float* g_v     = (const float*)d_in[19];
  const float* mixW    = (const float*)d_in[20];
  const float* mixb    = (const float*)d_in[21];
  const float* fc1     = (const float*)d_in[22];
  const float* fc2     = (const float*)d_in[23];
  const float* alpha   = (const float*)d_in[24];
  const float* beta    = (const float*)d_in[25];
  float* out = (float*)d_out;

  float* ws      = (float*)d_ws;        // ~4.06 MiB of fp32 scratch
  float* s_h     = ws;                  // NBC*HH
  float* s_w     = s_h + NBC * HH;      // NBC*WWD
  float* w_h     = s_w + NBC * WWD;     // NBC*HH
  float* w_w     = w_h + NBC * HH;      // NBC*WWD
  float* mean_bc = w_w + NBC * WWD;     // NBC
  float* max_bc  = mean_bc + NBC;       // NBC
  float* gh      = max_bc + NBC;        // NBC
  float* gw      = gh + NBC;            // NBC
  float* alh     = gw + NBC;            // NBC
  float* alw     = alh + NBC;           // NBC
  float* comb    = alw + NBC;           // NBC

  pool_kernel<<<NBC, 256, 0, stream>>>(x, hw_sq_w, hw_sq_b, ww_sq_w, ww_sq_b,
                                       s_h, s_w, mean_bc, max_bc);
  strip_kernel<<<NBC, 256, 0, stream>>>(s_h, s_w, hw_conv, ww_conv,
                                        hw_g, hw_b, hw_m, hw_v,
                                        ww_g, ww_b, ww_m, ww_v,
                                        gate_w, g_g, g_b, g_m, g_v,
                                        w_h, w_w, gh, gw);
  lam_kernel<<<(NBC + 255) / 256, 256, 0, stream>>>(gh, gw, mixW, mixb, alpha,
                                                    alh, alw);
  ca_kernel<<<1, 32, 0, stream>>>(mean_bc, max_bc, fc1, fc2, beta, comb);
  apply_kernel<<<NBC, 256, 0, stream>>>(x, w_h, w_w, alh, alw, comb, out);
}